// SimpleEquivariantLayer_23922967838730
// MI455X (gfx1250) — compile-verified
//
#include <hip/hip_runtime.h>

typedef __attribute__((ext_vector_type(2))) float v2f;
typedef __attribute__((ext_vector_type(8))) float v8f;

namespace {
constexpr int kN = 100000;   // nodes
constexpr int kE = 1250000;  // edges
constexpr int kF = 64;       // feature width (in = hid = out)
constexpr float kC0 = 0.28209479177387814f;
}  // namespace

// ---------------------------------------------------------------------------
// Zero the scatter accumulators (seg [N*64] and cnt [N], stored contiguously).
// ---------------------------------------------------------------------------
__global__ __launch_bounds__(256) void zero_kernel(float* __restrict__ p, int n) {
  int i = blockIdx.x * blockDim.x + threadIdx.x;
  const int stride = gridDim.x * blockDim.x;
  for (; i < n; i += stride) p[i] = 0.0f;
}

// ---------------------------------------------------------------------------
// Stage 1: xw = x @ (W_tp * scale)   using V_WMMA_F32_16X16X4_F32.
// One wave computes a 16x64 output tile; 8 waves (256 thr) -> 128 rows/block.
// scale is folded into the LDS copy of W_tp so the store loop is pure stores.
// A-frag: lane = M (lane&15), half-wave picks the K pair; D-frag:
// M = vgpr + 8*(lane>=16), N = lane&15 (ISA 7.12.2 layout tables).
// ---------------------------------------------------------------------------
__global__ __launch_bounds__(256) void tp_gemm_kernel(
    const float* __restrict__ x, const float* __restrict__ Wtp,
    float* __restrict__ xw, int n_rows, float scale) {
  __shared__ float sW[kF * kF];
  for (int i = threadIdx.x; i < kF * kF; i += 256) sW[i] = Wtp[i] * scale;
  __syncthreads();

  const int wave = threadIdx.x >> 5;
  const int lane = threadIdx.x & 31;
  const int hi = lane >> 4;   // half-wave: selects K pair
  const int ml = lane & 15;   // M for A-frag, N for B/D-frags
  const int row0 = blockIdx.x * 128 + wave * 16;
  const bool full = (row0 + 16 <= n_rows);  // wave-uniform fast path

  int arow = row0 + ml;
  if (arow >= n_rows) arow = n_rows - 1;  // clamp reads; writes are guarded
  const float* xr = x + arow * kF;

  v2f a[16];
#pragma unroll
  for (int kk = 0; kk < 16; ++kk) {
    const int k = kk * 4 + hi * 2;
    a[kk].x = xr[k];
    a[kk].y = xr[k + 1];
  }

#pragma unroll
  for (int nt = 0; nt < 4; ++nt) {
    const int n = nt * 16 + ml;
    v8f c = {};
#pragma unroll
    for (int kk = 0; kk < 16; ++kk) {
      const int k = kk * 4 + hi * 2;
      v2f b;
      b.x = sW[k * kF + n];
      b.y = sW[(k + 1) * kF + n];
      c = __builtin_amdgcn_wmma_f32_16x16x4_f32(false, a[kk], false, b,
                                                (short)0, c, false, false);
    }
    float* wp = xw + (row0 + hi * 8) * kF + n;
    if (full) {
#pragma unroll
      for (int r = 0; r < 8; ++r) wp[r * kF] = c[r];
    } else {
#pragma unroll
      for (int r = 0; r < 8; ++r) {
        if (row0 + hi * 8 + r < n_rows) wp[r * kF] = c[r];
      }
    }
  }
}

// ---------------------------------------------------------------------------
// Stage 2: per-edge gather + scatter-add. One wave per edge: lane l handles
// channels {2l, 2l+1}. xw (25.6 MB) and seg (25.6 MB) are L2-resident on
// MI455X (192 MB L2), so the random traffic stays on-die; the adds are L2
// f32 atomics. Lane 0 also bumps the destination count.
// ---------------------------------------------------------------------------
__global__ __launch_bounds__(256) void edge_scatter_kernel(
    const int* __restrict__ eidx, const float* __restrict__ xw,
    float* __restrict__ seg, float* __restrict__ cnt, int n_edges) {
  const int t = blockIdx.x * blockDim.x + threadIdx.x;  // < 40M, fits i32
  const int e = t >> 5;
  const int l = t & 31;
  if (e >= n_edges) return;
  const int dst = eidx[e];            // row = destination segment
  const int src = eidx[n_edges + e];  // col = gather source
  const v2f v = *(const v2f*)(xw + src * kF + l * 2);  // global_load_b64
  float* sp = seg + dst * kF + l * 2;
  atomicAdd(sp, v.x);
  atomicAdd(sp + 1, v.y);
  if (l == 0) atomicAdd(cnt + dst, 1.0f);
}

// ---------------------------------------------------------------------------
// Stage 3 (fused): aggr = seg / max(cnt,1); h = relu(aggr@W1 + b1);
// out = h@W2 + b2.  4 waves (128 thr) -> 64 rows/block. W1+W2 in LDS (32 KB),
// h staged through a per-wave 16x64 LDS slab purely to convert the WMMA
// D-layout of GEMM1 into the A-layout needed by GEMM2.
// ---------------------------------------------------------------------------
__global__ __launch_bounds__(128) void mlp_kernel(
    const float* __restrict__ seg, const float* __restrict__ cnt,
    const float* __restrict__ W1, const float* __restrict__ b1,
    const float* __restrict__ W2, const float* __restrict__ b2,
    float* __restrict__ out, int n_rows) {
  __shared__ float sW1[kF * kF];
  __shared__ float sW2[kF * kF];
  __shared__ float sH[4][16 * kF];
  for (int i = threadIdx.x; i < kF * kF; i += 128) {
    sW1[i] = W1[i];
    sW2[i] = W2[i];
  }
  __syncthreads();

  const int wave = threadIdx.x >> 5;
  const int lane = threadIdx.x & 31;
  const int hi = lane >> 4;
  const int ml = lane & 15;
  const int row0 = blockIdx.x * 64 + wave * 16;
  const bool full = (row0 + 16 <= n_rows);  // wave-uniform fast path

  int arow = row0 + ml;
  if (arow >= n_rows) arow = n_rows - 1;
  const float inv = 1.0f / fmaxf(cnt[arow], 1.0f);
  const float* sr = seg + arow * kF;

  v2f a[16];
#pragma unroll
  for (int kk = 0; kk < 16; ++kk) {
    const int k = kk * 4 + hi * 2;
    a[kk].x = sr[k] * inv;
    a[kk].y = sr[k + 1] * inv;
  }

  // --- GEMM1: h = relu(aggr @ W1 + b1), staged to LDS in row-major form ---
#pragma unroll
  for (int nt = 0; nt < 4; ++nt) {
    const int n = nt * 16 + ml;
    const float bb = b1[n];
    v8f c = {bb, bb, bb, bb, bb, bb, bb, bb};
#pragma unroll
    for (int kk = 0; kk < 16; ++kk) {
      const int k = kk * 4 + hi * 2;
      v2f b;
      b.x = sW1[k * kF + n];
      b.y = sW1[(k + 1) * kF + n];
      c = __builtin_amdgcn_wmma_f32_16x16x4_f32(false, a[kk], false, b,
                                                (short)0, c, false, false);
    }
#pragma unroll
    for (int r = 0; r < 8; ++r)
      sH[wave][(r + hi * 8) * kF + n] = fmaxf(c[r], 0.0f);
  }
  __syncthreads();  // D-layout -> A-layout shuffle visibility

  v2f ah[16];
#pragma unroll
  for (int kk = 0; kk < 16; ++kk) {
    const int k = kk * 4 + hi * 2;
    ah[kk].x = sH[wave][ml * kF + k];
    ah[kk].y = sH[wave][ml * kF + k + 1];
  }

  // --- GEMM2: out = h @ W2 + b2 ---
#pragma unroll
  for (int nt = 0; nt < 4; ++nt) {
    const int n = nt * 16 + ml;
    const float bb = b2[n];
    v8f c = {bb, bb, bb, bb, bb, bb, bb, bb};
#pragma unroll
    for (int kk = 0; kk < 16; ++kk) {
      const int k = kk * 4 + hi * 2;
      v2f b;
      b.x = sW2[k * kF + n];
      b.y = sW2[(k + 1) * kF + n];
      c = __builtin_amdgcn_wmma_f32_16x16x4_f32(false, ah[kk], false, b,
                                                (short)0, c, false, false);
    }
    float* op = out + (row0 + hi * 8) * kF + n;
    if (full) {
#pragma unroll
      for (int r = 0; r < 8; ++r) op[r * kF] = c[r];
    } else {
#pragma unroll
      for (int r = 0; r < 8; ++r) {
        if (row0 + hi * 8 + r < n_rows) op[r * kF] = c[r];
      }
    }
  }
}

// ---------------------------------------------------------------------------
extern "C" void kernel_launch(void* const* d_in, const int* in_sizes, int n_in,
                              void* d_out, int out_size, void* d_ws,
                              size_t ws_size, hipStream_t stream) {
  (void)in_sizes; (void)n_in; (void)out_size; (void)ws_size;
  const float* x = (const float*)d_in[0];
  const int* eidx = (const int*)d_in[1];
  // d_in[2] = pos: unused — only the constant Y0 (0e) channel couples to the
  // all-scalar tp output, so positions cannot affect the result.
  const float* Wtp = (const float*)d_in[3];
  const float* W1 = (const float*)d_in[4];
  const float* b1 = (const float*)d_in[5];
  const float* W2 = (const float*)d_in[6];
  const float* b2 = (const float*)d_in[7];
  float* out = (float*)d_out;

  float* xw = (float*)d_ws;                    // [N, 64]
  float* seg = xw + (size_t)kN * kF;           // [N, 64]
  float* cnt = seg + (size_t)kN * kF;          // [N]

  const float scale = kC0 / 8.0f;  // C0 / sqrt(in_features=64)

  zero_kernel<<<2048, 256, 0, stream>>>(seg, kN * kF + kN);
  tp_gemm_kernel<<<(kN + 127) / 128, 256, 0, stream>>>(x, Wtp, xw, kN, scale);
  const long scat_threads = (long)kE * 32;
  edge_scatter_kernel<<<(int)((scat_threads + 255) / 256), 256, 0, stream>>>(
      eidx, xw, seg, cnt, kE);
  mlp_kernel<<<(kN + 63) / 64, 128, 0, stream>>>(seg, cnt, W1, b1, W2, b2, out,
                                                 kN);
}